// E_01_HSE_85942295593529
// MI455X (gfx1250) — compile-verified
//
#include <hip/hip_runtime.h>

typedef float v2f __attribute__((ext_vector_type(2)));
typedef float v8f __attribute__((ext_vector_type(8)));
typedef unsigned int u32x4 __attribute__((ext_vector_type(4)));
typedef unsigned int u32x8 __attribute__((ext_vector_type(8)));

namespace {
constexpr int kL    = 4096;    // sequence length
constexpr int kC    = 16;      // channels
constexpr int kPC   = 8;       // patch channels
constexpr int kD    = 1024;    // hidden dim
constexpr int kFEAT = 1024;    // feature dim
constexpr int kM    = 16384;   // B*P rows
constexpr int MT    = 32;      // rows per workgroup
constexpr int KT    = 8;       // K tile depth
// TDM pads 4 dwords after every 256 dwords streamed -> effective row stride:
constexpr int BROW  = kD + 16; // 1040 floats; 2 rows apart == +32 banks (conflict-free frags)
}

// Raw 32-bit LDS byte offset of a shared-memory pointer (generic -> AS3 -> int).
__device__ __forceinline__ unsigned lds_bytes(const void* p) {
    return (unsigned)(unsigned long long)(__attribute__((address_space(3))) const void*)p;
}

// One TDM DMA: W tile (KT rows x 1024 f32, row stride 1024) -> LDS with padding.
// D# layout per CDNA5 ISA (groups 0/1; 2D tensor -> groups 2/3 NULL).
__device__ __forceinline__ void tdm_load_b_tile(const float* gsrc, unsigned lds_byte_addr) {
    const unsigned long long ga = (unsigned long long)gsrc;
    u32x4 g0;
    g0[0] = 1u;                                          // count=1 (valid user descriptor)
    g0[1] = lds_byte_addr;                               // lds_addr (bytes)
    g0[2] = (unsigned)(ga & 0xffffffffu);                // global_addr[31:0]
    g0[3] = (unsigned)((ga >> 32) & 0x01ffffffu)         // global_addr[56:32]
          | 0x80000000u;                                 // type = 2 ("image") in [127:126]
    u32x8 g1;
    g1[0] = (2u << 16)                                   // data_size = 4 bytes
          | (1u << 20)                                   // pad_enable
          | (7u << 22)                                   // pad_interval: every 256 dwords
          | (3u << 25);                                  // pad_amount: 4 dwords
    g1[1] = ((unsigned)kD) << 16;                        // tensor_dim0 = 1024 (low 16)
    g1[2] = ((unsigned)KT) << 16;                        // tensor_dim0 hi=0 | tensor_dim1 = KT
    g1[3] = ((unsigned)kD) << 16;                        // tensor_dim1 hi=0 | tile_dim0 = 1024
    g1[4] = (unsigned)KT;                                // tile_dim1 = KT, tile_dim2 = 0
    g1[5] = (unsigned)kD;                                // tensor_dim0_stride = 1024 (low 32)
    g1[6] = 0u;                                          // stride0 hi | dim1_stride lo
    g1[7] = 0u;                                          // dim1_stride hi
    asm volatile("tensor_load_to_lds %0, %1" :: "s"(g0), "s"(g1) : "memory");
}

__device__ __forceinline__ void wait_tdm() {
#if __has_builtin(__builtin_amdgcn_s_wait_tensorcnt)
    __builtin_amdgcn_s_wait_tensorcnt(0);
#else
    asm volatile("s_wait_tensorcnt 0x0" ::: "memory");
#endif
}

// FIRST:  A = gathered patch features, epilogue = bias + SiLU, writes h -> hbuf
// !FIRST: A = h rows from hbuf, epilogue = bias, writes out -> hbuf in place
//         (rows are block-exclusive; all reads of them precede the final barrier).
template <bool FIRST>
__global__ __launch_bounds__(256)
void mlp_gemm_wmma(const float* __restrict__ x,
                   const float* __restrict__ W,
                   const float* __restrict__ bias,
                   const int*   __restrict__ startL,
                   const int*   __restrict__ startC,
                   const int*   __restrict__ fsPtr,
                   float* hbuf)
{
    __shared__ float sB[2][KT * BROW];    // 2 x 32.5 KB, TDM-filled, padded rows
    __shared__ float sA[2][MT][KT + 2];   // row stride 10 -> 8B-aligned K-pairs, no conflicts
    __shared__ int   sSL[MT];
    __shared__ int   sSC[MT];

    const int tid  = threadIdx.x;
    const int blk  = blockIdx.x;
    const int row0 = blk * MT;

    float T = 0.0f;
    if (FIRST) {
        T = 1.0f / (float)fsPtr[0];
        if (tid < MT) {
            sSL[tid] = startL[row0 + tid];
            sSC[tid] = startC[row0 + tid];
        }
        __syncthreads();
    }

    const int wave  = tid >> 5;
    const int lane  = tid & 31;
    const int m16   = lane & 15;
    const int kb    = (lane >> 4) << 1;     // K-pair of this half-wave: 0 or 2
    const int nbase = wave * 128;           // 8 waves x 128 output columns
    const int arow  = tid >> 3;             // A staging: row 0..31
    const int akq   = tid & 7;              // A staging: k 0..7
    const bool issuer = (wave == 0);        // one wave issues TDM per workgroup

    auto stageA = [&](int kk, int buf) {
        float av;
        if (FIRST) {
            // feature f = l*16 + c ; c<8 -> x[b,(sl+l)%L,(sc+c)%C] ; c>=8 -> t[(sl+l)%L]
            const int f    = kk + akq;
            const int l    = f >> 4;
            const int c    = f & 15;
            const int lidx = (sSL[arow] + l) & (kL - 1);
            if (c < kPC) {
                const int cidx = (sSC[arow] + c) & (kC - 1);
                const int b    = (row0 + arow) >> 8;
                av = x[((size_t)b * kL + lidx) * kC + cidx];
            } else {
                av = (float)lidx * T;
            }
        } else {
            av = hbuf[(size_t)(row0 + arow) * kD + (kk + akq)];
        }
        sA[buf][arow][akq] = av;
    };

    v8f acc[2][8];
    #pragma unroll
    for (int mt = 0; mt < 2; ++mt)
        #pragma unroll
        for (int nt = 0; nt < 8; ++nt)
            #pragma unroll
            for (int j = 0; j < 8; ++j)
                acc[mt][nt][j] = 0.0f;

    // ---- prologue: fill buffer 0 ----
    if (issuer) tdm_load_b_tile(W, lds_bytes(&sB[0][0]));
    stageA(0, 0);
    if (issuer) wait_tdm();
    __syncthreads();

    // ---- main loop: TDM(next) overlaps WMMA(current) ----
    for (int kk = 0; kk < kFEAT; kk += KT) {
        const int cur = (kk / KT) & 1;
        const int nxt = cur ^ 1;
        if (kk + KT < kFEAT) {
            if (issuer)
                tdm_load_b_tile(W + (size_t)(kk + KT) * kD, lds_bytes(&sB[nxt][0]));
            stageA(kk + KT, nxt);
        }

        const float* bb = &sB[cur][0];
        #pragma unroll
        for (int ks = 0; ks < KT; ks += 4) {
            v2f a0, a1;
            a0[0] = sA[cur][m16     ][ks + kb];
            a0[1] = sA[cur][m16     ][ks + kb + 1];
            a1[0] = sA[cur][m16 + 16][ks + kb];
            a1[1] = sA[cur][m16 + 16][ks + kb + 1];
            #pragma unroll
            for (int nt = 0; nt < 8; ++nt) {
                const int col  = nbase + nt * 16 + m16;
                const int bidx = (ks + kb) * BROW + col + ((col >> 8) << 2); // TDM pad skew
                v2f bf;
                bf[0] = bb[bidx];
                bf[1] = bb[bidx + BROW];
                acc[0][nt] = __builtin_amdgcn_wmma_f32_16x16x4_f32(
                    false, a0, false, bf, (short)0, acc[0][nt], false, false);
                acc[1][nt] = __builtin_amdgcn_wmma_f32_16x16x4_f32(
                    false, a1, false, bf, (short)0, acc[1][nt], false, false);
            }
        }

        if (issuer) wait_tdm();   // next tile's DMA done (overlapped with compute above)
        __syncthreads();
    }

    // ---- epilogue: C/D layout -> VGPR j holds M = j + 8*(lane>>4), N = lane&15 ----
    const int mhalf = (lane >> 4) << 3;
    #pragma unroll
    for (int mt = 0; mt < 2; ++mt) {
        #pragma unroll
        for (int nt = 0; nt < 8; ++nt) {
            const int n   = nbase + nt * 16 + m16;
            const float bv = bias[n];
            #pragma unroll
            for (int j = 0; j < 8; ++j) {
                const int m = mt * 16 + mhalf + j;
                float v = acc[mt][nt][j] + bv;
                if (FIRST) v = v / (1.0f + __expf(-v));   // SiLU
                hbuf[(size_t)(row0 + m) * kD + n] = v;
            }
        }
    }
}

extern "C" void kernel_launch(void* const* d_in, const int* in_sizes, int n_in,
                              void* d_out, int out_size, void* d_ws, size_t ws_size,
                              hipStream_t stream) {
    const float* x  = (const float*)d_in[0];
    const float* W1 = (const float*)d_in[1];
    const float* b1 = (const float*)d_in[2];
    const float* W2 = (const float*)d_in[3];
    const float* b2 = (const float*)d_in[4];
    const int*   sL = (const int*)d_in[5];
    const int*   sC = (const int*)d_in[6];
    const int*   fs = (const int*)d_in[7];
    float* out = (float*)d_out;   // h scratch for layer 1, overwritten in place by layer 2

    dim3 grid(kM / MT), block(256);
    mlp_gemm_wmma<true ><<<grid, block, 0, stream>>>(x, W1, b1, sL, sC, fs, out);
    mlp_gemm_wmma<false><<<grid, block, 0, stream>>>(x, W2, b2, sL, sC, fs, out);
}